// SlaterPooling_54743653154930
// MI455X (gfx1250) — compile-verified
//
#include <hip/hip_runtime.h>
#include <hip/hip_bf16.h>

typedef float v2f __attribute__((ext_vector_type(2)));
typedef float v8f __attribute__((ext_vector_type(8)));

#define LDA 33   // 32x32 tile with +1 padding -> conflict-free column walks

// One wave32 factors one gathered 32x32 matrix held in LDS and returns det.
// Blocked LUP: [16-col panel w/ partial pivoting] -> [U12 trsm] ->
// [A22 -= L21*U12 via 4x v_wmma_f32_16x16x4_f32] -> [trailing 16x16 LUP].
__device__ __forceinline__ float det32_wave(const float* __restrict__ rows, // &mo[b][spin0][0], row stride 64
                                            const int* __restrict__ cols,  // 32 sorted MO indices
                                            float* __restrict__ A)         // 32*LDA floats LDS
{
    const int lane = threadIdx.x;          // blockDim.x == 32
    const int cj = cols[lane];             // column gathered by this lane

    // ---- Gather: A[r][lane] = mo[b, r, cols[lane]] (L2-resident) ----
    #pragma unroll 4
    for (int r = 0; r < 32; ++r)
        A[r * LDA + lane] = rows[r * 64 + cj];
    __syncthreads();

    float det = 1.0f;                      // uniform across lanes

    // ---- Phase A: panel LU, k = 0..15, pivot among rows k..31, swap full rows ----
    for (int k = 0; k < 16; ++k) {
        // cross-lane argmax of |A[i][k]| for i >= k (lane i holds row i)
        float av = (lane >= k) ? fabsf(A[lane * LDA + k]) : -1.0f;
        int   p  = lane;
        #pragma unroll
        for (int off = 16; off > 0; off >>= 1) {
            float av2 = __shfl_xor(av, off, 32);
            int   p2  = __shfl_xor(p,  off, 32);
            if (av2 > av || (av2 == av && p2 < p)) { av = av2; p = p2; }
        }
        if (p != k) {                      // uniform branch (p uniform)
            det = -det;                    // lane j swaps column j of rows k,p
            float tk = A[k * LDA + lane];
            float tp = A[p * LDA + lane];
            A[k * LDA + lane] = tp;
            A[p * LDA + lane] = tk;
        }
        __syncthreads();
        float piv = A[k * LDA + k];        // broadcast load
        det *= piv;
        float rp = 1.0f / piv;
        if (lane > k) {                    // lane = row i, update panel cols only
            float l = A[lane * LDA + k] * rp;
            A[lane * LDA + k] = l;         // store multiplier (needed for L11/L21)
            for (int j = k + 1; j < 16; ++j)
                A[lane * LDA + j] -= l * A[k * LDA + j];
        }
        __syncthreads();
    }

    // ---- Phase B: U12 = L11^{-1} * A12 (rows 0..15, cols 16..31) ----
    {
        const int jj   = 16 + (lane & 15); // column handled by this lane
        const int half = lane >> 4;        // split rows across the two lane halves
        for (int k = 0; k < 15; ++k) {
            float ukj = A[k * LDA + jj];
            for (int i = k + 1 + half; i < 16; i += 2)
                A[i * LDA + jj] -= A[i * LDA + k] * ukj;
            __syncthreads();
        }
    }

    // ---- Phase C: A22 -= L21 * U12  (16x16x16 via 4x WMMA f32 16x16x4) ----
    {
        const int n  = lane & 15;
        const int hi = lane >> 4;          // 0 or 1
        v8f c;
        #pragma unroll
        for (int r = 0; r < 8; ++r)        // C/D layout: M = r + 8*hi, N = n
            c[r] = A[(16 + r + 8 * hi) * LDA + 16 + n];
        #pragma unroll
        for (int t = 0; t < 4; ++t) {
            const int kb = 4 * t + 2 * hi; // A layout: lanes 16-31 carry K+2
            v2f av, bv;
            av.x = -A[(16 + n) * LDA + kb];        // -L21[M=n][kb]
            av.y = -A[(16 + n) * LDA + kb + 1];    // -L21[M=n][kb+1]
            bv.x =  A[kb       * LDA + 16 + n];    //  U12[kb][N=n]
            bv.y =  A[(kb + 1) * LDA + 16 + n];    //  U12[kb+1][N=n]
            c = __builtin_amdgcn_wmma_f32_16x16x4_f32(
                    false, av, false, bv, (short)0, c, false, false);
        }
        __syncthreads();                   // all reads of A22/L21/U12 done
        #pragma unroll
        for (int r = 0; r < 8; ++r)
            A[(16 + r + 8 * hi) * LDA + 16 + n] = c[r];
    }
    __syncthreads();

    // ---- Phase D: factor trailing 16x16, pivot among rows 16..31 ----
    for (int k = 16; k < 32; ++k) {
        float av = (lane >= k) ? fabsf(A[lane * LDA + k]) : -1.0f;
        int   p  = lane;
        #pragma unroll
        for (int off = 16; off > 0; off >>= 1) {
            float av2 = __shfl_xor(av, off, 32);
            int   p2  = __shfl_xor(p,  off, 32);
            if (av2 > av || (av2 == av && p2 < p)) { av = av2; p = p2; }
        }
        if (p != k) {
            det = -det;
            float tk = A[k * LDA + lane];
            float tp = A[p * LDA + lane];
            A[k * LDA + lane] = tp;
            A[p * LDA + lane] = tk;
        }
        __syncthreads();
        float piv = A[k * LDA + k];
        det *= piv;
        float rp = 1.0f / piv;
        if (lane > k) {
            float l = A[lane * LDA + k] * rp;  // multiplier not needed later
            for (int j = k + 1; j < 32; ++j)
                A[lane * LDA + j] -= l * A[k * LDA + j];
        }
        __syncthreads();
    }
    return det;   // sign * product of all 32 pivots, uniform across lanes
}

__global__ __launch_bounds__(32)
void slater_pool_kernel(const float* __restrict__ mo,
                        const int*   __restrict__ cfg_up,
                        const int*   __restrict__ cfg_dn,
                        float*       __restrict__ out)
{
    __shared__ float A[32 * LDA];          // 4224 B: up/dn reuse the same tile
    const int wg = blockIdx.x;             // 0 .. NB*NCONF-1
    const int b  = wg >> 7;                // / 128
    const int c  = wg & 127;               // % 128
    const float* base = mo + (size_t)b * 64 * 64;

    float du = det32_wave(base,            cfg_up + c * 32, A);   // electrons 0..31
    __syncthreads();
    float dd = det32_wave(base + 32 * 64,  cfg_dn + c * 32, A);   // electrons 32..63

    if (threadIdx.x == 0)
        out[wg] = du * dd;                 // out[b*128 + c]
}

extern "C" void kernel_launch(void* const* d_in, const int* in_sizes, int n_in,
                              void* d_out, int out_size, void* d_ws, size_t ws_size,
                              hipStream_t stream)
{
    const float* mo  = (const float*)d_in[0];   // (1024, 64, 64) f32
    const int*   cup = (const int*)  d_in[1];   // (128, 32) i32
    const int*   cdn = (const int*)  d_in[2];   // (128, 32) i32
    float*       out = (float*)d_out;           // (1024, 128) f32

    const int nblocks = 1024 * 128;             // one wave32 per (b, config)
    slater_pool_kernel<<<nblocks, 32, 0, stream>>>(mo, cup, cdn, out);
}